// TransitionDown_15539191676963
// MI455X (gfx1250) — compile-verified
//
#include <hip/hip_runtime.h>

// ---------- problem constants ----------
#define NPTS   100000
#define NSAMP  16384
#define KNN    16
#define CH     128
#define MROWS  (NSAMP * KNN)        // 262144
#define BN_EPS 1e-5f

// ---------- WMMA types (fp32 16x16x4) ----------
typedef __attribute__((ext_vector_type(2))) float v2f;
typedef __attribute__((ext_vector_type(8))) float v8f;

// ---------- TDM descriptor types (6-arg clang-23 builtin) ----------
typedef __attribute__((ext_vector_type(4))) unsigned int v4u;
typedef __attribute__((ext_vector_type(8))) int          v8i;
typedef __attribute__((ext_vector_type(4))) int          v4i;

// DMA one contiguous row of nElems f32 from global -> LDS via the Tensor
// Data Mover.  Descriptor layout per CDNA5 ISA:
//  g0: [1:0]=count=1, [63:32]=lds_addr, [120:64]=global_addr, [127:126]=type=2
//  g1: [17:16]=data_size(2 -> 4B), [79:48]=tensor_dim0, [111:80]=tensor_dim1,
//      [127:112]=tile_dim0, [143:128]=tile_dim1, [207:160]=tensor_dim0_stride
__device__ __forceinline__ void tdm_load_row_f32(unsigned ldsOff,
                                                 const void* gptr,
                                                 unsigned nElems) {
    unsigned long long ga = (unsigned long long)(size_t)gptr;
    v4u g0;
    g0[0] = 1u;                                            // count = 1 (user D#)
    g0[1] = ldsOff;                                        // lds_addr (bytes)
    g0[2] = (unsigned)(ga & 0xffffffffu);                  // global_addr[31:0]
    g0[3] = (unsigned)((ga >> 32) & 0x1ffffffu) | (2u << 30); // addr[56:32] | type=2
    v8i g1;
    g1[0] = (int)(2u << 16);                               // data_size = 2 (4 bytes)
    g1[1] = (int)((nElems & 0xffffu) << 16);               // tensor_dim0[15:0]
    g1[2] = (int)(((nElems >> 16) & 0xffffu) | (1u << 16));// tensor_dim0[31:16], tensor_dim1=1
    g1[3] = (int)((nElems & 0xffffu) << 16);               // tile_dim0 = nElems
    g1[4] = 1;                                             // tile_dim1 = 1, tile_dim2 = 0
    g1[5] = (int)nElems;                                   // tensor_dim0_stride[31:0]
    g1[6] = 0;                                             // stride hi / dim1_stride lo
    g1[7] = 0;
    v4i g2 = {0, 0, 0, 0};
    v4i g3 = {0, 0, 0, 0};
    v8i g4 = {0, 0, 0, 0, 0, 0, 0, 0};
    __builtin_amdgcn_tensor_load_to_lds(g0, g1, g2, g3, g4, 0);
}

// =====================================================================
// Init: reset grid-sync state, seed FPS with index 0
// =====================================================================
__global__ void init_kernel(unsigned* syncv, float* lastPt, int* sampled,
                            const float* __restrict__ points) {
    if (threadIdx.x == 0 && blockIdx.x == 0) {
        syncv[0] = 0u; syncv[1] = 0u;
        lastPt[0] = points[0]; lastPt[1] = points[1]; lastPt[2] = points[2];
        sampled[0] = 0;
    }
}

// =====================================================================
// FPS: persistent kernel, 64 blocks x 256 threads = 16384 threads,
// each owns <=7 points fully in registers. Two grid syncs per step.
// =====================================================================
#define FPS_BLOCKS 64
#define FPS_NT     (FPS_BLOCKS * 256)

__device__ __forceinline__ void gsync(unsigned* sv, unsigned sidx) {
    __syncthreads();
    if (threadIdx.x == 0) {
        __threadfence();
        unsigned prev = atomicAdd(&sv[0], 1u);
        if (prev == sidx * (unsigned)FPS_BLOCKS + (FPS_BLOCKS - 1u)) {
            atomicExch(&sv[1], sidx + 1u);
        }
        while (atomicAdd(&sv[1], 0u) <= sidx) { __builtin_amdgcn_s_sleep(2); }
    }
    __syncthreads();
}

__global__ __launch_bounds__(256) void fps_kernel(
    const float* __restrict__ points, unsigned* syncv,
    volatile float* lastPt, float* partD, int* partI, int* sampled) {
    const int tid = threadIdx.x;
    const int gid = blockIdx.x * 256 + tid;

    float px[7], py[7], pz[7], md[7];
#pragma unroll
    for (int j = 0; j < 7; ++j) {
        int i = gid + j * FPS_NT;
        if (i < NPTS) {
            px[j] = points[i * 3 + 0];
            py[j] = points[i * 3 + 1];
            pz[j] = points[i * 3 + 2];
            md[j] = 1e10f;
        } else {
            px[j] = 0.f; py[j] = 0.f; pz[j] = 0.f;
            md[j] = -3.4e38f;   // never selected, min() keeps it
        }
    }

    __shared__ float sd[256];
    __shared__ int   si[256];
    unsigned sidx = 0;

    for (int t = 1; t < NSAMP; ++t) {
        float lx = lastPt[0], ly = lastPt[1], lz = lastPt[2];
        float bd = -3.4e38f; int bi = 0x7fffffff;
#pragma unroll
        for (int j = 0; j < 7; ++j) {
            float dx = px[j] - lx, dy = py[j] - ly, dz = pz[j] - lz;
            float d  = dx * dx + dy * dy + dz * dz;
            d = fminf(md[j], d);
            md[j] = d;
            int i = gid + j * FPS_NT;
            if (d > bd || (d == bd && i < bi)) { bd = d; bi = i; }
        }
        sd[tid] = bd; si[tid] = bi;
        __syncthreads();
        for (int s = 128; s > 0; s >>= 1) {
            if (tid < s) {
                float d2 = sd[tid + s]; int i2 = si[tid + s];
                if (d2 > sd[tid] || (d2 == sd[tid] && i2 < si[tid])) {
                    sd[tid] = d2; si[tid] = i2;
                }
            }
            __syncthreads();
        }
        if (tid == 0) { partD[blockIdx.x] = sd[0]; partI[blockIdx.x] = si[0]; }
        gsync(syncv, sidx); ++sidx;

        if (blockIdx.x == 0 && tid == 0) {
            volatile float* pD = partD; volatile int* pI = partI;
            float gbd = -3.4e38f; int gbi = 0x7fffffff;
            for (int b = 0; b < FPS_BLOCKS; ++b) {
                float d2 = pD[b]; int i2 = pI[b];
                if (d2 > gbd || (d2 == gbd && i2 < gbi)) { gbd = d2; gbi = i2; }
            }
            sampled[t] = gbi;
            lastPt[0] = points[gbi * 3 + 0];
            lastPt[1] = points[gbi * 3 + 1];
            lastPt[2] = points[gbi * 3 + 2];
        }
        gsync(syncv, sidx); ++sidx;
    }
}

// =====================================================================
// KNN: one thread per query, LDS-tiled points (float4: xyz + |p|^2),
// register-resident sorted top-16 with index tie-break.
// =====================================================================
#define KTILE 2048
__global__ __launch_bounds__(256) void knn_kernel(
    const float* __restrict__ points, const int* __restrict__ sampled,
    int* __restrict__ knn) {
    __shared__ float4 tile[KTILE];
    const int tid = threadIdx.x;
    const int s   = blockIdx.x * 256 + tid;      // 0..16383
    const int qi  = sampled[s];
    const float qx = points[qi * 3 + 0];
    const float qy = points[qi * 3 + 1];
    const float qz = points[qi * 3 + 2];
    const float q2 = qx * qx + qy * qy + qz * qz;

    float bd[16]; int bi[16];
#pragma unroll
    for (int j = 0; j < 16; ++j) { bd[j] = 3.4e38f; bi[j] = 0x7fffffff; }

    for (int base = 0; base < NPTS; base += KTILE) {
        int cnt = min(KTILE, NPTS - base);
        __syncthreads();
        for (int i = tid; i < cnt; i += 256) {
            float x = points[(base + i) * 3 + 0];
            float y = points[(base + i) * 3 + 1];
            float z = points[(base + i) * 3 + 2];
            tile[i] = make_float4(x, y, z, x * x + y * y + z * z);
        }
        __syncthreads();
        for (int i = 0; i < cnt; ++i) {
            float4 p = tile[i];
            float dot = qx * p.x + qy * p.y + qz * p.z;
            float d   = (q2 + p.w) - 2.0f * dot;   // same formula as reference
            int gidx  = base + i;
            if (d < bd[15] || (d == bd[15] && gidx < bi[15])) {
                float cd = d; int ci = gidx;
#pragma unroll
                for (int j = 0; j < 16; ++j) {
                    bool sw = (cd < bd[j]) || (cd == bd[j] && ci < bi[j]);
                    float td = bd[j]; int ti = bi[j];
                    if (sw) { bd[j] = cd; bi[j] = ci; cd = td; ci = ti; }
                }
            }
        }
    }
#pragma unroll
    for (int j = 0; j < 16; ++j) knn[s * 16 + j] = bi[j];
}

// =====================================================================
// GEMM1: Y1[r, :] = features[knn[r], :] @ W1  via V_WMMA_F32_16X16X4_F32.
// Block = 8 waves covering 128 rows x 128 cols; each wave = 64x32
// (4 row-tiles x 2 col-tiles, 8 accumulators).  W1 staged into LDS by
// the Tensor Data Mover (tensor_load_to_lds + s_wait_tensorcnt).
// =====================================================================
__global__ __launch_bounds__(256) void gemm1_kernel(
    const int* __restrict__ knn, const float* __restrict__ feat,
    const float* __restrict__ W, float* __restrict__ Y) {
    __shared__ float ldsW[CH * CH];                       // 64 KB

    if (threadIdx.x == 0) {
        tdm_load_row_f32((unsigned)(size_t)ldsW, W, CH * CH);
        __builtin_amdgcn_s_wait_tensorcnt((short)0);
    }
    __syncthreads();

    const int lane   = threadIdx.x & 31;
    const int wave   = threadIdx.x >> 5;
    const int rowGrp = wave >> 2;                         // 0..1 (64-row group)
    const int colGrp = wave & 3;                          // 0..3 (32-col group)
    const int l15    = lane & 15;
    const int koff   = (lane >> 4) << 1;                  // 0 or 2
    const int rowBlk = blockIdx.x * 128 + rowGrp * 64;
    const int nb0    = colGrp * 32 + l15;

    const float* arow[4];
#pragma unroll
    for (int rt = 0; rt < 4; ++rt)
        arow[rt] = feat + (long)knn[rowBlk + rt * 16 + l15] * CH;

    v8f acc[4][2];
#pragma unroll
    for (int rt = 0; rt < 4; ++rt) { acc[rt][0] = {}; acc[rt][1] = {}; }

    for (int k0 = 0; k0 < CH; k0 += 4) {
        const int ka = k0 + koff, kb = ka + 1;
        v2f a[4];
#pragma unroll
        for (int rt = 0; rt < 4; ++rt) { a[rt].x = arow[rt][ka]; a[rt].y = arow[rt][kb]; }
        v2f b0, b1;
        b0.x = ldsW[ka * CH + nb0];      b0.y = ldsW[kb * CH + nb0];
        b1.x = ldsW[ka * CH + nb0 + 16]; b1.y = ldsW[kb * CH + nb0 + 16];
#pragma unroll
        for (int rt = 0; rt < 4; ++rt) {
            acc[rt][0] = __builtin_amdgcn_wmma_f32_16x16x4_f32(false, a[rt], false, b0, (short)0, acc[rt][0], false, false);
            acc[rt][1] = __builtin_amdgcn_wmma_f32_16x16x4_f32(false, a[rt], false, b1, (short)0, acc[rt][1], false, false);
        }
    }

    const int mOff = (lane < 16) ? 0 : 8;
#pragma unroll
    for (int rt = 0; rt < 4; ++rt) {
#pragma unroll
        for (int j = 0; j < 8; ++j) {
            float* yr = Y + (long)(rowBlk + rt * 16 + mOff + j) * CH;
            yr[nb0]      = acc[rt][0][j];
            yr[nb0 + 16] = acc[rt][1][j];
        }
    }
}

// =====================================================================
// GEMM2: Y2 = relu(bn1(Y1)) @ W2 (BN fused into A-load; W2 + BN params
// staged into LDS; same 64x32-per-wave WMMA layout).
// =====================================================================
__global__ __launch_bounds__(256) void gemm2_kernel(
    const float* __restrict__ Y1, const float* __restrict__ sc,
    const float* __restrict__ sh, const float* __restrict__ W,
    float* __restrict__ Y) {
    __shared__ float ldsW[CH * CH];                       // 64 KB
    __shared__ float lsc[CH], lsh[CH];

    if (threadIdx.x == 0) {
        tdm_load_row_f32((unsigned)(size_t)ldsW, W, CH * CH);
        __builtin_amdgcn_s_wait_tensorcnt((short)0);
    }
    if (threadIdx.x < CH) {
        lsc[threadIdx.x] = sc[threadIdx.x];
        lsh[threadIdx.x] = sh[threadIdx.x];
    }
    __syncthreads();

    const int lane   = threadIdx.x & 31;
    const int wave   = threadIdx.x >> 5;
    const int rowGrp = wave >> 2;
    const int colGrp = wave & 3;
    const int l15    = lane & 15;
    const int koff   = (lane >> 4) << 1;
    const int rowBlk = blockIdx.x * 128 + rowGrp * 64;
    const int nb0    = colGrp * 32 + l15;

    const float* arow[4];
#pragma unroll
    for (int rt = 0; rt < 4; ++rt)
        arow[rt] = Y1 + (long)(rowBlk + rt * 16 + l15) * CH;

    v8f acc[4][2];
#pragma unroll
    for (int rt = 0; rt < 4; ++rt) { acc[rt][0] = {}; acc[rt][1] = {}; }

    for (int k0 = 0; k0 < CH; k0 += 4) {
        const int ka = k0 + koff, kb = ka + 1;
        const float sca = lsc[ka], sha = lsh[ka];
        const float scb = lsc[kb], shb = lsh[kb];
        v2f a[4];
#pragma unroll
        for (int rt = 0; rt < 4; ++rt) {
            a[rt].x = fmaxf(arow[rt][ka] * sca + sha, 0.f);
            a[rt].y = fmaxf(arow[rt][kb] * scb + shb, 0.f);
        }
        v2f b0, b1;
        b0.x = ldsW[ka * CH + nb0];      b0.y = ldsW[kb * CH + nb0];
        b1.x = ldsW[ka * CH + nb0 + 16]; b1.y = ldsW[kb * CH + nb0 + 16];
#pragma unroll
        for (int rt = 0; rt < 4; ++rt) {
            acc[rt][0] = __builtin_amdgcn_wmma_f32_16x16x4_f32(false, a[rt], false, b0, (short)0, acc[rt][0], false, false);
            acc[rt][1] = __builtin_amdgcn_wmma_f32_16x16x4_f32(false, a[rt], false, b1, (short)0, acc[rt][1], false, false);
        }
    }

    const int mOff = (lane < 16) ? 0 : 8;
#pragma unroll
    for (int rt = 0; rt < 4; ++rt) {
#pragma unroll
        for (int j = 0; j < 8; ++j) {
            float* yr = Y + (long)(rowBlk + rt * 16 + mOff + j) * CH;
            yr[nb0]      = acc[rt][0][j];
            yr[nb0 + 16] = acc[rt][1][j];
        }
    }
}

// =====================================================================
// Column stats: per-block coalesced partial (sum, sumsq) over 1024 rows
// =====================================================================
__global__ __launch_bounds__(256) void colpart_kernel(
    const float* __restrict__ Y, float* __restrict__ part, int rowsPerBlock) {
    __shared__ float ls[CH], ls2[CH];
    const int col  = threadIdx.x & 127;
    const int half = threadIdx.x >> 7;
    const long r0  = (long)blockIdx.x * rowsPerBlock + half;
    float s = 0.f, s2 = 0.f;
    for (int i = 0; i < rowsPerBlock; i += 2) {
        float v = Y[(r0 + i) * CH + col];
        s += v; s2 += v * v;
    }
    if (half) { ls[col] = s; ls2[col] = s2; }
    __syncthreads();
    if (!half) {
        s += ls[col]; s2 += ls2[col];
        part[(long)blockIdx.x * 256 + col]       = s;
        part[(long)blockIdx.x * 256 + 128 + col] = s2;
    }
}

__global__ void bnfinal_kernel(const float* __restrict__ part, int nparts,
                               const float* __restrict__ gamma,
                               const float* __restrict__ beta,
                               float* __restrict__ sc, float* __restrict__ sh,
                               float invM) {
    int c = threadIdx.x;   // 128 threads
    float s = 0.f, s2 = 0.f;
    for (int b = 0; b < nparts; ++b) {
        s  += part[(long)b * 256 + c];
        s2 += part[(long)b * 256 + 128 + c];
    }
    float mean = s * invM;
    float var  = s2 * invM - mean * mean;
    float a    = gamma[c] * rsqrtf(var + BN_EPS);
    sc[c] = a;
    sh[c] = beta[c] - mean * a;
}

// =====================================================================
// Finalize: relu(bn2(Y2)) max-pooled over K=16, write out_features
// =====================================================================
__global__ __launch_bounds__(256) void pool_kernel(
    const float* __restrict__ Y2, const float* __restrict__ sc,
    const float* __restrict__ sh, float* __restrict__ out) {
    int idx = blockIdx.x * 256 + threadIdx.x;    // 0 .. S*C-1
    int s = idx >> 7, c = idx & 127;
    float a = sc[c], b = sh[c];
    float m = -3.4e38f;
#pragma unroll
    for (int k = 0; k < KNN; ++k) {
        float v = Y2[((long)s * KNN + k) * CH + c];
        v = fmaxf(v * a + b, 0.f);
        m = fmaxf(m, v);
    }
    out[NSAMP * 3 + idx] = m;
}

__global__ void samp_kernel(const int* __restrict__ sampled,
                            const float* __restrict__ points,
                            float* __restrict__ out) {
    int i = blockIdx.x * 256 + threadIdx.x;
    if (i < NSAMP * 3) {
        int s = i / 3, d = i - s * 3;
        out[i] = points[(long)sampled[s] * 3 + d];
    }
}

// =====================================================================
// Host launch
// =====================================================================
extern "C" void kernel_launch(void* const* d_in, const int* in_sizes, int n_in,
                              void* d_out, int out_size, void* d_ws, size_t ws_size,
                              hipStream_t stream) {
    const float* points  = (const float*)d_in[0];   // [N,3]
    const float* feats   = (const float*)d_in[1];   // [N,C]
    const float* W1      = (const float*)d_in[2];   // [C,C]
    const float* gamma1  = (const float*)d_in[3];
    const float* beta1   = (const float*)d_in[4];
    const float* W2      = (const float*)d_in[5];
    const float* gamma2  = (const float*)d_in[6];
    const float* beta2   = (const float*)d_in[7];
    float* out = (float*)d_out;

    char* w = (char*)d_ws;
    unsigned* syncv  = (unsigned*)(w + 0);                 // 2 u32
    float*    lastPt = (float*)(w + 64);                   // 3 f32
    float*    partD  = (float*)(w + 128);                  // 64 f32
    int*      partI  = (int*)(w + 512);                    // 64 i32
    int*      sampled= (int*)(w + 1024);                   // 16384 i32
    int*      knn    = (int*)(w + 1024 + 65536);           // 262144 i32
    float*    bnbuf  = (float*)(w + 1115136);              // 4*128 f32
    float*    sc1 = bnbuf, *sh1 = bnbuf + 128, *sc2 = bnbuf + 256, *sh2 = bnbuf + 384;
    float*    colpart= (float*)(w + 1117184);              // 256*256 f32
    float*    Y1     = (float*)(w + 1379328);              // 262144*128 f32
    float*    Y2     = (float*)(w + 1379328 + (size_t)MROWS * CH * 4);

    // 1) init + FPS
    init_kernel<<<1, 64, 0, stream>>>(syncv, lastPt, sampled, points);
    fps_kernel<<<FPS_BLOCKS, 256, 0, stream>>>(points, syncv, lastPt, partD, partI, sampled);

    // 2) KNN
    knn_kernel<<<NSAMP / 256, 256, 0, stream>>>(points, sampled, knn);

    // 3) GEMM1 (gather fused) -> Y1
    gemm1_kernel<<<MROWS / 128, 256, 0, stream>>>(knn, feats, W1, Y1);

    // 4) BN1 stats
    colpart_kernel<<<256, 256, 0, stream>>>(Y1, colpart, MROWS / 256);
    bnfinal_kernel<<<1, 128, 0, stream>>>(colpart, 256, gamma1, beta1, sc1, sh1,
                                          1.0f / (float)MROWS);

    // 5) GEMM2 (BN1+ReLU fused into A) -> Y2
    gemm2_kernel<<<MROWS / 128, 256, 0, stream>>>(Y1, sc1, sh1, W2, Y2);

    // 6) BN2 stats
    colpart_kernel<<<256, 256, 0, stream>>>(Y2, colpart, MROWS / 256);
    bnfinal_kernel<<<1, 128, 0, stream>>>(colpart, 256, gamma2, beta2, sc2, sh2,
                                          1.0f / (float)MROWS);

    // 7) finalize: BN2+ReLU+maxpool + sampled_points
    pool_kernel<<<(NSAMP * CH) / 256, 256, 0, stream>>>(Y2, sc2, sh2, out);
    samp_kernel<<<(NSAMP * 3 + 255) / 256, 256, 0, stream>>>(sampled, points, out);
}